// GCN_1589137899719
// MI455X (gfx1250) — compile-verified
//
#include <hip/hip_runtime.h>

#define N_NODES   100000
#define N_EDGES   3200000
#define IN_FEAT   128
#define HIDDEN    5
#define N_CLASSES 16

typedef __attribute__((ext_vector_type(2))) float v2f;
typedef __attribute__((ext_vector_type(8))) float v8f;

// ---------------------------------------------------------------------------
// Degree / normalization
// ---------------------------------------------------------------------------
__global__ void k_init_deg(float* __restrict__ deg) {
    int i = blockIdx.x * blockDim.x + threadIdx.x;
    if (i < N_NODES) deg[i] = 1.0f;  // self-loop contributes 1
}

__global__ void k_deg_accum(const int* __restrict__ ei, float* __restrict__ deg) {
    int e = blockIdx.x * blockDim.x + threadIdx.x;
    if (e < N_EDGES) {
        int d = ei[N_EDGES + e];  // dst row of edge_index
        atomicAdd(&deg[d], 1.0f);
    }
}

__global__ void k_dinv(float* __restrict__ deg) {
    int i = blockIdx.x * blockDim.x + threadIdx.x;
    if (i < N_NODES) deg[i] = rsqrtf(deg[i]);  // deg >= 1 always
}

// ---------------------------------------------------------------------------
// h1 = x @ W1  via V_WMMA_F32_16X16X4_F32
// One wave per 16-node tile, 32 chained WMMAs over K=128.
// A layout (16x4 f32): lane L holds row M=L%16; lanes 0-15 carry K={k,k+1},
// lanes 16-31 carry K={k+2,k+3}. B mirrored (col = lane%16). C: 8 VGPRs,
// row = v + 8*(lane/16), col = lane%16.
// ---------------------------------------------------------------------------
__global__ void k_xw1_wmma(const float* __restrict__ x,
                           const float* __restrict__ W1,
                           float* __restrict__ h1) {
    __shared__ float sB[IN_FEAT * 16];  // W1 zero-padded to 16 columns
    for (int i = threadIdx.x; i < IN_FEAT * 16; i += blockDim.x) {
        int col = i & 15, row = i >> 4;
        sB[i] = (col < HIDDEN) ? W1[row * HIDDEN + col] : 0.0f;
    }
    __syncthreads();

    const int NT   = N_NODES / 16;  // 6250 tiles, exact
    int wave = (blockIdx.x * blockDim.x + threadIdx.x) >> 5;  // wave-uniform
    int lane = threadIdx.x & 31;
    if (wave >= NT) return;

    int m   = lane & 15;   // row (A) / column (B,C)
    int grp = lane >> 4;   // K-pair selector

    const float* xrow = x + (size_t)(wave * 16 + m) * IN_FEAT + grp * 2;

    v8f c = {0.f, 0.f, 0.f, 0.f, 0.f, 0.f, 0.f, 0.f};
#pragma unroll
    for (int k0 = 0; k0 < IN_FEAT; k0 += 4) {
        v2f a;
        a[0] = xrow[k0];
        a[1] = xrow[k0 + 1];
        v2f b;
        b[0] = sB[(k0 + grp * 2)     * 16 + m];
        b[1] = sB[(k0 + grp * 2 + 1) * 16 + m];
        c = __builtin_amdgcn_wmma_f32_16x16x4_f32(false, a, false, b,
                                                  (short)0, c, false, false);
    }

    if (m < HIDDEN) {
#pragma unroll
        for (int v = 0; v < 8; ++v) {
            int row = wave * 16 + v + grp * 8;
            h1[(size_t)row * HIDDEN + m] = c[v];
        }
    }
}

// ---------------------------------------------------------------------------
// Self-loop seed:  agg1[i] = h1[i] * dinv[i]^2
// ---------------------------------------------------------------------------
__global__ void k_selfloop1(const float* __restrict__ h1,
                            const float* __restrict__ dinv,
                            float* __restrict__ agg1) {
    int i = blockIdx.x * blockDim.x + threadIdx.x;
    if (i >= N_NODES) return;
    float w = dinv[i] * dinv[i];
#pragma unroll
    for (int f = 0; f < HIDDEN; ++f)
        agg1[(size_t)i * HIDDEN + f] = h1[(size_t)i * HIDDEN + f] * w;
}

// ---------------------------------------------------------------------------
// Edge scatter-add in hidden space (used for both layers).
// feat has row stride HIDDEN; agg has row stride outStride.
// ---------------------------------------------------------------------------
__global__ void k_edges(const int* __restrict__ ei,
                        const float* __restrict__ dinv,
                        const float* __restrict__ feat,
                        float* __restrict__ agg,
                        int outStride) {
    int e = blockIdx.x * blockDim.x + threadIdx.x;
    if (e >= N_EDGES) return;
    int s = ei[e];
    int d = ei[N_EDGES + e];
    __builtin_prefetch(&feat[(size_t)s * HIDDEN], 0, 0);
    float nrm = dinv[s] * dinv[d];
#pragma unroll
    for (int f = 0; f < HIDDEN; ++f)
        atomicAdd(&agg[(size_t)d * outStride + f],
                  feat[(size_t)s * HIDDEN + f] * nrm);
}

// ---------------------------------------------------------------------------
// a1 = relu(agg1 + b1); seed layer-2 aggregation (padded [N][8], cols 5..7 = 0)
// ---------------------------------------------------------------------------
__global__ void k_relu_seed(const float* __restrict__ agg1,
                            const float* __restrict__ b1,
                            const float* __restrict__ dinv,
                            float* __restrict__ a1,
                            float* __restrict__ agg2) {
    int i = blockIdx.x * blockDim.x + threadIdx.x;
    if (i >= N_NODES) return;
    float w = dinv[i] * dinv[i];
#pragma unroll
    for (int f = 0; f < HIDDEN; ++f) {
        float v = agg1[(size_t)i * HIDDEN + f] + b1[f];
        v = fmaxf(v, 0.0f);
        a1[(size_t)i * HIDDEN + f]   = v;
        agg2[(size_t)i * 8 + f]      = v * w;
    }
    agg2[(size_t)i * 8 + 5] = 0.0f;
    agg2[(size_t)i * 8 + 6] = 0.0f;
    agg2[(size_t)i * 8 + 7] = 0.0f;
}

// ---------------------------------------------------------------------------
// out = log_softmax(agg2 @ W2 + b2)  via WMMA (K padded 5->8) + shfl_xor
// reductions across the 16-lane column groups of the C tile.
// ---------------------------------------------------------------------------
__global__ void k_out_wmma(const float* __restrict__ agg2,
                           const float* __restrict__ W2,
                           const float* __restrict__ b2,
                           float* __restrict__ out) {
    __shared__ float sB[8 * 16];
    __shared__ float sb2[16];
    for (int i = threadIdx.x; i < 8 * 16; i += blockDim.x) {
        int col = i & 15, row = i >> 4;
        sB[i] = (row < HIDDEN) ? W2[row * N_CLASSES + col] : 0.0f;
    }
    if (threadIdx.x < 16) sb2[threadIdx.x] = b2[threadIdx.x];
    __syncthreads();

    const int NT   = N_NODES / 16;
    int wave = (blockIdx.x * blockDim.x + threadIdx.x) >> 5;
    int lane = threadIdx.x & 31;
    if (wave >= NT) return;

    int m   = lane & 15;
    int grp = lane >> 4;

    const float* arow = agg2 + (size_t)(wave * 16 + m) * 8 + grp * 2;

    v8f c = {0.f, 0.f, 0.f, 0.f, 0.f, 0.f, 0.f, 0.f};
#pragma unroll
    for (int k0 = 0; k0 < 8; k0 += 4) {
        v2f a;
        a[0] = arow[k0];
        a[1] = arow[k0 + 1];
        v2f b;
        b[0] = sB[(k0 + grp * 2)     * 16 + m];
        b[1] = sB[(k0 + grp * 2 + 1) * 16 + m];
        c = __builtin_amdgcn_wmma_f32_16x16x4_f32(false, a, false, b,
                                                  (short)0, c, false, false);
    }

    float bias = sb2[m];
#pragma unroll
    for (int v = 0; v < 8; ++v) {
        float z = c[v] + bias;
        // reductions stay within each 16-lane half (xor masks < 16)
        float mx = z;
#pragma unroll
        for (int off = 1; off < 16; off <<= 1)
            mx = fmaxf(mx, __shfl_xor(mx, off, 32));
        float ex = expf(z - mx);
        float sum = ex;
#pragma unroll
        for (int off = 1; off < 16; off <<= 1)
            sum += __shfl_xor(sum, off, 32);
        float res = z - mx - logf(sum);
        int row = wave * 16 + v + grp * 8;
        out[(size_t)row * N_CLASSES + m] = res;
    }
}

// ---------------------------------------------------------------------------
extern "C" void kernel_launch(void* const* d_in, const int* in_sizes, int n_in,
                              void* d_out, int out_size, void* d_ws, size_t ws_size,
                              hipStream_t stream) {
    const float* x  = (const float*)d_in[0];
    const int*   ei = (const int*)d_in[1];   // [2, E]: src row then dst row
    const float* W1 = (const float*)d_in[2];
    const float* b1 = (const float*)d_in[3];
    const float* W2 = (const float*)d_in[4];
    const float* b2 = (const float*)d_in[5];
    float* out = (float*)d_out;

    float* ws   = (float*)d_ws;
    float* deg  = ws;                                  // N   (becomes dinv)
    float* h1   = deg  + N_NODES;                      // N*5 (reused as a1)
    float* agg1 = h1   + (size_t)N_NODES * HIDDEN;     // N*5
    float* agg2 = agg1 + (size_t)N_NODES * HIDDEN;     // N*8 padded

    const int TPB        = 256;
    const int nodeBlocks = (N_NODES + TPB - 1) / TPB;
    const int edgeBlocks = (N_EDGES + TPB - 1) / TPB;
    const int tileBlocks = (N_NODES / 16 + 7) / 8;     // 8 waves (wave32) / block

    k_init_deg <<<nodeBlocks, TPB, 0, stream>>>(deg);
    k_deg_accum<<<edgeBlocks, TPB, 0, stream>>>(ei, deg);
    k_dinv     <<<nodeBlocks, TPB, 0, stream>>>(deg);

    k_xw1_wmma <<<tileBlocks, TPB, 0, stream>>>(x, W1, h1);
    k_selfloop1<<<nodeBlocks, TPB, 0, stream>>>(h1, deg, agg1);
    k_edges    <<<edgeBlocks, TPB, 0, stream>>>(ei, deg, h1, agg1, HIDDEN);

    k_relu_seed<<<nodeBlocks, TPB, 0, stream>>>(agg1, b1, deg, h1 /*a1*/, agg2);
    k_edges    <<<edgeBlocks, TPB, 0, stream>>>(ei, deg, h1 /*a1*/, agg2, 8);

    k_out_wmma <<<tileBlocks, TPB, 0, stream>>>(agg2, W2, b2, out);
}